// TransCell_81037442941613
// MI455X (gfx1250) — compile-verified
//
#include <hip/hip_runtime.h>
#include <hip/hip_bf16.h>

#define B_    128
#define N_    36
#define HID_  1024
#define EDGE_ 512
#define NN_   (N_ * N_)          // 1296
#define NEGV  (-1e12f)

typedef __attribute__((ext_vector_type(2))) float v2f;
typedef __attribute__((ext_vector_type(8))) float v8f;

// ---------------------------------------------------------------------------
// Kernel 1: w[b,e] = relu( (q @ Wq^T)[b,e] + bq[e] ) * Wo[e]
// WMMA f32 16x16x4, one wave per 16x16 output tile, K = HID.
//   A = q        [B, HID]   (M = batch, K = hid)
//   B = Wq^T     [HID, E]   => B[k][n] = Wq[n*HID + k]
// A layout (32-bit, 16x4): lanes 0-15 hold M=lane, VGPR0=K0, VGPR1=K1;
//                          lanes 16-31 hold M=lane-16, VGPR0=K2, VGPR1=K3.
// => each lane loads a contiguous f32 pair at k0 + 2*(lane>=16).
// C/D layout: VGPR v: lanes 0-15 -> M=v, N=lane; lanes 16-31 -> M=v+8.
// ---------------------------------------------------------------------------
__global__ void __launch_bounds__(128)
gemm_w_kernel(const float* __restrict__ q,
              const float* __restrict__ Wq,
              const float* __restrict__ bq,
              const float* __restrict__ Wo,
              float* __restrict__ w)
{
    const int wave = blockIdx.x * (blockDim.x >> 5) + (threadIdx.x >> 5);
    const int lane = threadIdx.x & 31;
    const int ntiles = EDGE_ / 16;          // 32
    const int mt = wave / ntiles;           // 0..7
    const int nt = wave % ntiles;           // 0..31
    const int m0 = mt * 16;
    const int n0 = nt * 16;
    const int half = lane >> 4;             // 0 or 1
    const int ln   = lane & 15;

    const float* arow = q  + (size_t)(m0 + ln) * HID_ + half * 2;
    const float* brow = Wq + (size_t)(n0 + ln) * HID_ + half * 2;

    v8f c = {};
    #pragma unroll 4
    for (int k0 = 0; k0 < HID_; k0 += 4) {
        v2f a = *(const v2f*)(arow + k0);
        v2f b = *(const v2f*)(brow + k0);
        c = __builtin_amdgcn_wmma_f32_16x16x4_f32(
                /*neg_a=*/false, a, /*neg_b=*/false, b,
                /*c_mod=*/(short)0, c, /*reuse_a=*/false, /*reuse_b=*/false);
    }

    const int col = n0 + ln;
    const float bias = bq[col];
    const float wo   = Wo[col];
    #pragma unroll
    for (int v = 0; v < 8; ++v) {
        const int row = m0 + v + half * 8;
        float val = c[v] + bias;
        val = val > 0.f ? val : 0.f;
        w[(size_t)row * EDGE_ + col] = val * wo;
    }
}

// ---------------------------------------------------------------------------
// Kernel 2: relation[b,i,j] = mask ? relu( dot(trans_mat[b,i,j,:], w[b,:]) + bo )
//                                  : -1e12
// Bandwidth-bound streaming pass over 340 MB. One wave per (i,j) row,
// 8 waves/block, w[b] staged in LDS. Each lane: 4x float4 loads (E=512 exact).
// ---------------------------------------------------------------------------
__global__ void __launch_bounds__(256)
relation_kernel(const float* __restrict__ tm,     // [B, NN, E]
                const float* __restrict__ w,      // [B, E]
                const int*   __restrict__ rmask,  // [B, NN]
                const float* __restrict__ bo,     // [1]
                float* __restrict__ rel)          // [B, NN]
{
    __shared__ float ws[EDGE_];

    const int blocks_per_b = NN_ / 8;             // 162
    const int b    = blockIdx.x / blocks_per_b;
    const int rblk = blockIdx.x % blocks_per_b;

    for (int i = threadIdx.x; i < EDGE_; i += blockDim.x)
        ws[i] = w[(size_t)b * EDGE_ + i];
    __syncthreads();

    const int wave = threadIdx.x >> 5;
    const int lane = threadIdx.x & 31;
    const int row  = rblk * 8 + wave;             // 0..1295
    const size_t ridx = (size_t)b * NN_ + row;
    const float* rp = tm + ridx * EDGE_;

    // gfx1250 prefetch of the back half of this row
    __builtin_prefetch(rp + 256, 0, 0);

    float acc = 0.f;
    #pragma unroll
    for (int c = 0; c < 4; ++c) {
        const int e = c * 128 + lane * 4;
        const float4 t = *(const float4*)(rp + e);
        acc = fmaf(t.x, ws[e + 0], acc);
        acc = fmaf(t.y, ws[e + 1], acc);
        acc = fmaf(t.z, ws[e + 2], acc);
        acc = fmaf(t.w, ws[e + 3], acc);
    }

    // wave32 butterfly reduction
    #pragma unroll
    for (int off = 16; off > 0; off >>= 1)
        acc += __shfl_xor(acc, off, 32);

    if (lane == 0) {
        float val = acc + bo[0];
        val = val > 0.f ? val : 0.f;
        rel[ridx] = (rmask[ridx] == 0) ? NEGV : val;
    }
}

// ---------------------------------------------------------------------------
// Kernel 3: softmax over axis=1 (over i, per column j), then
//           logits[b,i] = sum_j softmax(rel)[b,i,j] * z[b,j].
// One block per batch element; 36x36 tile in LDS.
// ---------------------------------------------------------------------------
__global__ void __launch_bounds__(64)
softmax_logits_kernel(const float* __restrict__ rel,  // [B, N, N]
                      const float* __restrict__ z,    // [B, N]
                      float* __restrict__ out)        // [B, N]
{
    __shared__ float R[NN_];
    __shared__ float colmax[N_];
    __shared__ float colsum[N_];

    const int b = blockIdx.x;
    const float* rb = rel + (size_t)b * NN_;

    for (int i = threadIdx.x; i < NN_; i += blockDim.x)
        R[i] = rb[i];
    __syncthreads();

    const int t = threadIdx.x;
    if (t < N_) {
        float m = -1e30f;
        #pragma unroll
        for (int i = 0; i < N_; ++i) m = fmaxf(m, R[i * N_ + t]);
        float s = 0.f;
        #pragma unroll
        for (int i = 0; i < N_; ++i) s += __expf(R[i * N_ + t] - m);
        colmax[t] = m;
        colsum[t] = s;
    }
    __syncthreads();

    if (t < N_) {
        float acc = 0.f;
        #pragma unroll
        for (int j = 0; j < N_; ++j) {
            const float p = __expf(R[t * N_ + j] - colmax[j]) / colsum[j];
            acc += p * z[(size_t)b * N_ + j];
        }
        out[(size_t)b * N_ + t] = acc;
    }
}

// ---------------------------------------------------------------------------
// Host-side launch
// Inputs (setup_inputs order): q, trans_mat, r_mask, z_logits, Wq, bq, Wo, bo
// ---------------------------------------------------------------------------
extern "C" void kernel_launch(void* const* d_in, const int* in_sizes, int n_in,
                              void* d_out, int out_size, void* d_ws, size_t ws_size,
                              hipStream_t stream) {
    const float* q    = (const float*)d_in[0];
    const float* tm   = (const float*)d_in[1];
    const int*   rm   = (const int*)  d_in[2];
    const float* z    = (const float*)d_in[3];
    const float* Wq   = (const float*)d_in[4];
    const float* bq   = (const float*)d_in[5];
    const float* Wo   = (const float*)d_in[6];   // [1, E] -> flat E
    const float* bo   = (const float*)d_in[7];
    float* out = (float*)d_out;

    float* w   = (float*)d_ws;                              // B*E   = 256 KB
    float* rel = (float*)((char*)d_ws + (size_t)B_ * EDGE_ * sizeof(float)); // B*NN

    // 1) WMMA GEMM + fused epilogue: 256 tiles, 4 waves (128 thr) per block.
    {
        const int tiles = (B_ / 16) * (EDGE_ / 16);         // 256 waves
        dim3 grid(tiles / 4), block(128);
        gemm_w_kernel<<<grid, block, 0, stream>>>(q, Wq, bq, Wo, w);
    }

    // 2) Streaming relation pass: 8 rows per block.
    {
        dim3 grid(B_ * (NN_ / 8)), block(256);              // 20736 blocks
        relation_kernel<<<grid, block, 0, stream>>>(tm, w, rm, bo, rel);
    }

    // 3) Softmax (axis=1) + logits.
    {
        dim3 grid(B_), block(64);
        softmax_logits_kernel<<<grid, block, 0, stream>>>(rel, z, out);
    }
}